// SoftDTW_75900662055047
// MI455X (gfx1250) — compile-verified
//
#include <hip/hip_runtime.h>

#define TT 512          // sequence length (fixed by reference)
#define CH 16           // diagonal rows per lane (32 lanes * 16 = 512)
#define BIGF 1e30f      // finite stand-in for +inf; exactly preserved by the
                        // boundary cells since ulp(1e30) >> |dd - log2(3)|
#define LOG2E_F 1.4426950408889634f
#define LN2_F   0.6931471805599453f

// Raw CDNA5 transcendentals: v_exp_f32 computes 2^x, v_log_f32 computes log2.
#if __has_builtin(__builtin_amdgcn_exp2f)
#define EXP2F(x) __builtin_amdgcn_exp2f(x)
#else
#define EXP2F(x) exp2f(x)
#endif
#if __has_builtin(__builtin_amdgcn_logf)
#define LOG2F(x) __builtin_amdgcn_logf(x)
#else
#define LOG2F(x) log2f(x)
#endif

// Shuffle-up by 1 lane within the wave32 (lane L gets lane L-1's value).
// Lane 0 result is a don't-care (overridden by boundary logic).
__device__ __forceinline__ float shfl_up1(float v) {
    int lane = (int)threadIdx.x;              // blockDim.x == 32 (one wave)
    int src = (lane == 0) ? 0 : (lane - 1);
    int r = __builtin_amdgcn_ds_bpermute(src << 2, __float_as_int(v));
    return __int_as_float(r);
}

// softmin in the log2-scaled domain (all DP values pre-scaled by log2(e)):
//   softmin'(a,b,c) = m - log2(1 + 2^(m-mid) + 2^(m-mx))
// using an exact 3-sort (fmin/fmax only) so the exp2(0)=1 term of the min
// element is folded to a constant: 2 TRANS exps instead of 3, balancing the
// TRANS pipe against co-executing VALU. m <= mid <= mx exactly, so the exp2
// arguments are always <= 0 (no overflow path).
__device__ __forceinline__ float softmin3_2(float a, float b, float c) {
    float lo  = fminf(a, b);
    float hi  = fmaxf(a, b);
    float m   = fminf(lo, c);                 // min
    float mid = fminf(hi, fmaxf(lo, c));      // median (exact)
    float mx  = fmaxf(hi, c);                 // max
    float s = 1.0f + EXP2F(m - mid) + EXP2F(m - mx);
    return m - LOG2F(s);
}

// One anti-diagonal step: A1 holds diag K-1, A2 holds diag K-2 and is
// overwritten IN PLACE with diag K. Iterate r descending so A2[r-1] (the
// diagonal neighbor) is still the old value when cell r is computed, and so
// the ds_bpermute results (only needed by r==0) are consumed last, hiding
// their latency behind 15 independent softmin evaluations.
// No validity masking: j<=0 boundary cells stay exactly BIGF (rounding
// argument above); j>512 positions hold garbage that is never read.
#define SDTW_STEP(A1, A2)                                                    \
    {                                                                        \
        float p1 = shfl_up1((A1)[CH - 1]);   /* R[i-1][j]   on diag K-1 */   \
        float p2 = shfl_up1((A2)[CH - 1]);   /* R[i-1][j-1] on diag K-2 */   \
        if (lane == 0) { p1 = BIGF; p2 = BIGF; } /* row-0 boundary, K>=3 */  \
        _Pragma("unroll")                                                    \
        for (int r = CH - 1; r >= 0; --r) {                                  \
            float up   = (r == 0) ? p1 : (A1)[r - 1];                        \
            float left = (A1)[r];                                            \
            float dg   = (r == 0) ? p2 : (A2)[r - 1];                        \
            (A2)[r] = dd[r] + softmin3_2(up, left, dg);                      \
        }                                                                    \
    }

// One wave32 per batch element. Anti-diagonal wavefront DP held entirely in
// VGPRs; ~1.5 ds_bpermute per diagonal step are the only cross-lane traffic.
__global__ __launch_bounds__(32, 1)
void sdtw_wave_kernel(const float* __restrict__ x,
                      const float* __restrict__ y,
                      float* __restrict__ ws) {
    const int b    = (int)blockIdx.x;
    const int lane = (int)threadIdx.x;        // 0..31
    const float* __restrict__ xb = x + b * TT;
    const float* __restrict__ yb = y + b * TT;

    // dd[r] = log2e * d[i-1] for i = lane*CH + r + 1  ->  log2e*(x-y)^2
    float dd[CH];
    {
        const float4* x4 = (const float4*)xb + lane * (CH / 4);
        const float4* y4 = (const float4*)yb + lane * (CH / 4);
#pragma unroll
        for (int q = 0; q < CH / 4; ++q) {
            float4 xv = x4[q];
            float4 yv = y4[q];
            float e0 = xv.x - yv.x, e1 = xv.y - yv.y;
            float e2 = xv.z - yv.z, e3 = xv.w - yv.w;
            dd[q * 4 + 0] = (e0 * e0) * LOG2E_F;
            dd[q * 4 + 1] = (e1 * e1) * LOG2E_F;
            dd[q * 4 + 2] = (e2 * e2) * LOG2E_F;
            dd[q * 4 + 3] = (e3 * e3) * LOG2E_F;
        }
    }

    // a1 = diag 1 (all boundary/out-of-range -> BIGF).
    float a1[CH], a2[CH];
#pragma unroll
    for (int r = 0; r < CH; ++r) a1[r] = BIGF;

    // Peel k = 2: only cell (1,1) exists:
    //   R'[1][1] = d'[0] + softmin'(INF, INF, R[0][0]=0) = d'[0].
    // a2 <- diag 2.
#pragma unroll
    for (int r = 0; r < CH; ++r) a2[r] = BIGF;
    if (lane == 0) a2[0] = dd[0];

    // Steady state: diagonals 3..2*TT (1022 steps), two per iteration with
    // the diagonal buffers ping-ponging in place (no rotation movs).
    for (int it = 0; it < TT - 1; ++it) {     // 511 iterations x 2 steps
        SDTW_STEP(a2, a1);                    // a1 <- next diagonal
        SDTW_STEP(a1, a2);                    // a2 <- next diagonal
    }

    // Last write was diag 2*TT into a2; R'[TT][TT] is lane 31, r = CH-1.
    // Un-scale back to nats: R = R' * ln(2).
    if (lane == 31) ws[b] = a2[CH - 1] * LN2_F;
}

// Deterministic scalar mean over B per-batch results (no atomics: safe for
// graph replay since d_out/d_ws are not re-poisoned between replays).
__global__ void sdtw_mean_kernel(const float* __restrict__ ws,
                                 float* __restrict__ out, int B) {
    if (blockIdx.x == 0 && threadIdx.x == 0) {
        float s = 0.0f;
        for (int i = 0; i < B; ++i) s += ws[i];
        out[0] = s / (float)B;
    }
}

extern "C" void kernel_launch(void* const* d_in, const int* in_sizes, int n_in,
                              void* d_out, int out_size, void* d_ws, size_t ws_size,
                              hipStream_t stream) {
    const float* x = (const float*)d_in[0];   // (B, T) float32
    const float* y = (const float*)d_in[1];   // (B, T) float32
    float* out = (float*)d_out;               // scalar float32
    float* ws  = (float*)d_ws;                // B floats of scratch

    const int B = in_sizes[0] / TT;           // 32 for the reference setup

    sdtw_wave_kernel<<<B, 32, 0, stream>>>(x, y, ws);
    sdtw_mean_kernel<<<1, 32, 0, stream>>>(ws, out, B);
}